// LIFSpike_88098369175794
// MI455X (gfx1250) — compile-verified
//
#include <hip/hip_runtime.h>
#include <cstdint>

// ---------------------------------------------------------------------------
// LIF spike scan over time axis (last, contiguous, T=32).
//   u_t = TAU * u_{t-1} * (1 - o_{t-1}) + x_t ;  o_t = (u_t > VTH) ? 1 : 0
// Memory-bound (268 MB traffic, ~134 MFLOP -> HBM roofline ~11.5 us).
// Stage tiles through LDS with gfx1250 async global<->LDS DMA so both global
// loads and stores are perfectly coalesced (consecutive lanes -> consecutive
// 16B), while the serial 32-step recurrence runs per-thread against a
// bank-conflict-free padded LDS row.
// ---------------------------------------------------------------------------

#define LIF_T    32
#define LIF_TAU  0.25f
#define LIF_VTH  0.5f
#define NPB      256               // neurons per block == threads per block
#define ROWF     (LIF_T + 4)       // padded LDS row stride (36 floats = 144 B)
#define CHUNKS   (LIF_T / 4)       // 8 x 16-byte chunks per neuron

#if __has_builtin(__builtin_amdgcn_global_load_async_to_lds_b128) && \
    __has_builtin(__builtin_amdgcn_global_store_async_from_lds_b128)
#define LIF_HAVE_ASYNC 1
#else
#define LIF_HAVE_ASYNC 0
#endif

#if LIF_HAVE_ASYNC
// Builtin params are typed vector pointers:
//   param0: int __attribute__((vector_size(16))) AS(1)*   (global side)
//   param1: the AS(3) LDS analogue
typedef int lif_v4i __attribute__((ext_vector_type(4)));
typedef __attribute__((address_space(1))) lif_v4i* lif_gv4p;
typedef __attribute__((address_space(3))) lif_v4i* lif_lv4p;

static __device__ __forceinline__ void g2l_b128(const void* g, void* l) {
  __builtin_amdgcn_global_load_async_to_lds_b128(
      (lif_gv4p)(unsigned long long)g,
      (lif_lv4p)(unsigned)(unsigned long long)l, 0, 0);
}
static __device__ __forceinline__ void l2g_b128(void* g, const void* l) {
  __builtin_amdgcn_global_store_async_from_lds_b128(
      (lif_gv4p)(unsigned long long)g,
      (lif_lv4p)(unsigned)(unsigned long long)l, 0, 0);
}
static __device__ __forceinline__ void wait_async0() {
#if __has_builtin(__builtin_amdgcn_s_wait_asynccnt)
  __builtin_amdgcn_s_wait_asynccnt(0);
#else
  asm volatile("s_wait_asynccnt 0" ::: "memory");
#endif
}
#endif

__global__ __launch_bounds__(NPB)
void lif_scan_kernel(const float* __restrict__ x, float* __restrict__ y) {
  __shared__ float tile[NPB * ROWF];          // 36 KB
  const int tid = threadIdx.x;
  const size_t chunk0 = (size_t)blockIdx.x * (size_t)(NPB * CHUNKS);

  // ---- stage-in: flat chunk mapping, consecutive lanes -> consecutive 16 B
#pragma unroll
  for (int i = 0; i < CHUNKS; ++i) {
    const int k = i * NPB + tid;
    const int n = k >> 3;                      // neuron within tile
    const int c = k & 7;                       // 16B chunk within neuron row
    const float* g = x + (chunk0 + (size_t)k) * 4u;
    float* l = &tile[n * ROWF + c * 4];
#if LIF_HAVE_ASYNC
    g2l_b128(g, l);
#else
    *(float4*)l = *(const float4*)g;
#endif
  }
#if LIF_HAVE_ASYNC
  wait_async0();
#endif
  __syncthreads();

  // ---- serial LIF recurrence, one neuron per thread, row lives in LDS.
  // Row stride 36 floats => lane l hits banks {36*l+4c .. +3} mod 64:
  // 16 disjoint 4-bank groups per half-wave -> conflict-free b128 accesses.
  float u = 0.0f, o = 0.0f;
  float4* row = (float4*)&tile[tid * ROWF];
#pragma unroll
  for (int c = 0; c < CHUNKS; ++c) {
    const float4 v = row[c];
    float4 r;
    u = LIF_TAU * u * (1.0f - o) + v.x; o = (u > LIF_VTH) ? 1.0f : 0.0f; r.x = o;
    u = LIF_TAU * u * (1.0f - o) + v.y; o = (u > LIF_VTH) ? 1.0f : 0.0f; r.y = o;
    u = LIF_TAU * u * (1.0f - o) + v.z; o = (u > LIF_VTH) ? 1.0f : 0.0f; r.z = o;
    u = LIF_TAU * u * (1.0f - o) + v.w; o = (u > LIF_VTH) ? 1.0f : 0.0f; r.w = o;
    row[c] = r;   // own row only: no cross-thread hazard before the barrier
  }
  __syncthreads();

  // ---- stage-out: same flat mapping, coalesced async LDS -> global
#pragma unroll
  for (int i = 0; i < CHUNKS; ++i) {
    const int k = i * NPB + tid;
    const int n = k >> 3;
    const int c = k & 7;
    float* g = y + (chunk0 + (size_t)k) * 4u;
    const float* l = &tile[n * ROWF + c * 4];
#if LIF_HAVE_ASYNC
    l2g_b128(g, l);
#else
    *(float4*)g = *(const float4*)l;
#endif
  }
#if LIF_HAVE_ASYNC
  wait_async0();
#endif
}

// Generic tail path (only used if neuron count is not a multiple of NPB).
__global__ void lif_tail_kernel(const float* __restrict__ x,
                                float* __restrict__ y, int n0, int N) {
  const int n = n0 + blockIdx.x * blockDim.x + threadIdx.x;
  if (n >= N) return;
  const float* xi = x + (size_t)n * LIF_T;
  float* yi = y + (size_t)n * LIF_T;
  float u = 0.0f, o = 0.0f;
#pragma unroll
  for (int t = 0; t < LIF_T; ++t) {
    u = LIF_TAU * u * (1.0f - o) + xi[t];
    o = (u > LIF_VTH) ? 1.0f : 0.0f;
    yi[t] = o;
  }
}

extern "C" void kernel_launch(void* const* d_in, const int* in_sizes, int n_in,
                              void* d_out, int out_size, void* d_ws, size_t ws_size,
                              hipStream_t stream) {
  (void)n_in; (void)out_size; (void)d_ws; (void)ws_size;
  const float* x = (const float*)d_in[0];
  float* y = (float*)d_out;

  const int total = in_sizes[0];           // 16*128*512*32
  const int N = total / LIF_T;             // neurons = 1,048,576
  const int blocks = N / NPB;              // 4096
  if (blocks > 0) {
    lif_scan_kernel<<<blocks, NPB, 0, stream>>>(x, y);
  }
  const int done = blocks * NPB;
  if (done < N) {
    const int rem = N - done;
    lif_tail_kernel<<<(rem + 255) / 256, 256, 0, stream>>>(x, y, done, N);
  }
}